// RetNetCLM_51548197486855
// MI455X (gfx1250) — compile-verified
//
#include <hip/hip_runtime.h>
#include <math.h>

// ---------------------------------------------------------------------------
// RetNet complex CLM forward on gfx1250 (MI455X), fp32 WMMA (16x16x4) path.
// Activations stored planar (separate real/imag float planes); weights read
// directly from complex64 (interleaved float2) and de-interleaved into LDS.
// ---------------------------------------------------------------------------

typedef float v2f __attribute__((ext_vector_type(2)));
typedef float v8f __attribute__((ext_vector_type(8)));

#define L_     4
#define H_     1024
#define F_     2048
#define HEADS_ 8
#define V_     32000
#define B_     2
#define S_     1024
#define HD_    128
#define MROWS_ (B_ * S_)
#define EPS_   1e-8f

__device__ __forceinline__ v8f wmma4(v2f a, v2f b, v8f c) {
  // D = A(16x4) * B(4x16) + C(16x16), fp32. 8-arg form:
  // (neg_a, A, neg_b, B, c_mod, C, reuse_a, reuse_b)
  return __builtin_amdgcn_wmma_f32_16x16x4_f32(false, a, false, b,
                                               (short)0, c, false, false);
}

// --------------------------- embedding gather ------------------------------
__global__ __launch_bounds__(256) void embed_kernel(
    const int* __restrict__ ids, const float* __restrict__ emb,
    float* __restrict__ Xr, float* __restrict__ Xi, int n) {
  int i = blockIdx.x * blockDim.x + threadIdx.x;
  if (i >= n) return;
  int bs = i >> 10;         // / H_
  int c  = i & (H_ - 1);
  Xr[i] = emb[(long)ids[bs] * H_ + c];
  Xi[i] = 0.f;
}

// ------------------------ complex layer norm (H) ---------------------------
__global__ __launch_bounds__(256) void cnorm_kernel(
    const float* __restrict__ Xr, const float* __restrict__ Xi,
    const float2* __restrict__ g, const float2* __restrict__ bb,
    float* __restrict__ Or, float* __restrict__ Oi, int Hn) {
  __shared__ float red[256];
  int row = blockIdx.x, tid = threadIdx.x;
  float xr[4], xi[4];
  int cnt = 0;
  float sr = 0.f, si = 0.f;
  for (int c = tid; c < Hn; c += 256) {
    float r_ = Xr[(long)row * Hn + c];
    float i_ = Xi[(long)row * Hn + c];
    xr[cnt] = r_; xi[cnt] = i_; ++cnt;
    sr += r_; si += i_;
  }
  red[tid] = sr; __syncthreads();
  for (int s = 128; s > 0; s >>= 1) { if (tid < s) red[tid] += red[tid + s]; __syncthreads(); }
  float mr = red[0] / (float)Hn; __syncthreads();
  red[tid] = si; __syncthreads();
  for (int s = 128; s > 0; s >>= 1) { if (tid < s) red[tid] += red[tid + s]; __syncthreads(); }
  float mi = red[0] / (float)Hn; __syncthreads();
  float v = 0.f;
  for (int k = 0; k < cnt; ++k) {
    float dr = xr[k] - mr, di = xi[k] - mi;
    v += dr * dr + di * di;
  }
  red[tid] = v; __syncthreads();
  for (int s = 128; s > 0; s >>= 1) { if (tid < s) red[tid] += red[tid + s]; __syncthreads(); }
  float inv = rsqrtf(red[0] / (float)Hn + EPS_);
  cnt = 0;
  for (int c = tid; c < Hn; c += 256) {
    float dr = (xr[cnt] - mr) * inv;
    float di = (xi[cnt] - mi) * inv;
    ++cnt;
    float2 gg = g[c], bv = bb[c];
    Or[(long)row * Hn + c] = gg.x * dr - gg.y * di + bv.x;
    Oi[(long)row * Hn + c] = gg.x * di + gg.y * dr + bv.y;
  }
}

// ------------------- per-head group norm (HD) + affine ---------------------
__global__ __launch_bounds__(128) void gnorm_kernel(
    const float* __restrict__ Yhr, const float* __restrict__ Yhi,
    const float2* __restrict__ g, const float2* __restrict__ bb,
    float* __restrict__ Or, float* __restrict__ Oi) {
  __shared__ float red[128];
  int rid = blockIdx.x;
  long bs = rid / HEADS_;
  int h = rid % HEADS_;
  int c = threadIdx.x;
  long off = bs * H_ + (long)h * HD_ + c;
  float r_ = Yhr[off], i_ = Yhi[off];
  red[c] = r_; __syncthreads();
  for (int s = 64; s > 0; s >>= 1) { if (c < s) red[c] += red[c + s]; __syncthreads(); }
  float mr = red[0] / (float)HD_; __syncthreads();
  red[c] = i_; __syncthreads();
  for (int s = 64; s > 0; s >>= 1) { if (c < s) red[c] += red[c + s]; __syncthreads(); }
  float mi = red[0] / (float)HD_; __syncthreads();
  float dr = r_ - mr, di = i_ - mi;
  red[c] = dr * dr + di * di; __syncthreads();
  for (int s = 64; s > 0; s >>= 1) { if (c < s) red[c] += red[c + s]; __syncthreads(); }
  float inv = rsqrtf(red[0] / (float)HD_ + EPS_);
  float yr = dr * inv, yi = di * inv;
  float2 gg = g[h * HD_ + c], bv = bb[h * HD_ + c];
  Or[off] = gg.x * yr - gg.y * yi + bv.x;
  Oi[off] = gg.x * yi + gg.y * yr + bv.y;
}

// --------------------- swish gate: R = G*sig(G)*Yg -------------------------
__global__ __launch_bounds__(256) void gate_kernel(
    const float* __restrict__ Gr, const float* __restrict__ Gi,
    const float* __restrict__ Ygr, const float* __restrict__ Ygi,
    float* __restrict__ Rr, float* __restrict__ Ri, int n) {
  int i = blockIdx.x * blockDim.x + threadIdx.x;
  if (i >= n) return;
  float gr = Gr[i], gi = Gi[i];
  // sig(z) = 1 / (1 + exp(-z));  exp(-z) = e^-x (cos y - i sin y)
  float ex = __expf(-gr);
  float wr = 1.f + ex * __cosf(gi);
  float wi = -ex * __sinf(gi);
  float dnm = wr * wr + wi * wi;
  float sr = wr / dnm, si = -wi / dnm;
  float tr = gr * sr - gi * si;
  float ti = gr * si + gi * sr;
  float yr = Ygr[i], yi = Ygi[i];
  Rr[i] = tr * yr - ti * yi;
  Ri[i] = tr * yi + ti * yr;
}

// ------------------ complex gelu: h *= sig(1.702 h) ------------------------
__global__ __launch_bounds__(256) void cgelu_kernel(
    float* __restrict__ Hr, float* __restrict__ Hi, int n) {
  int i = blockIdx.x * blockDim.x + threadIdx.x;
  if (i >= n) return;
  float x = Hr[i], y = Hi[i];
  float ex = __expf(-1.702f * x);
  float wr = 1.f + ex * __cosf(1.702f * y);
  float wi = -ex * __sinf(1.702f * y);
  float dnm = wr * wr + wi * wi;
  float sr = wr / dnm, si = -wi / dnm;
  Hr[i] = x * sr - y * si;
  Hi[i] = x * si + y * sr;
}

// ------------------------------ residual add -------------------------------
__global__ __launch_bounds__(256) void cadd_kernel(
    const float* __restrict__ Ar, const float* __restrict__ Ai,
    const float* __restrict__ Br2, const float* __restrict__ Bi2,
    float* __restrict__ Cr, float* __restrict__ Ci, int n) {
  int i = blockIdx.x * blockDim.x + threadIdx.x;
  if (i >= n) return;
  Cr[i] = Ar[i] + Br2[i];
  Ci[i] = Ai[i] + Bi2[i];
}

// -------------------------- complex GEMM (WMMA) ----------------------------
// C[M x N] = A[M x K] (planar) * B[K x N] (interleaved complex64)
// Tile: 128 rows (8 waves x 16) x 64 cols, K staged 16-wide through LDS.
#define GM 128
#define GN 64
#define GK 16
#define AP 20   // 16 + 4 pad (bank-spread)
#define BP 68   // 64 + 4 pad

__global__ __launch_bounds__(256) void cgemm_kernel(
    const float* __restrict__ Ar, const float* __restrict__ Ai,
    const float2* __restrict__ Bc,
    float* __restrict__ Cr, float* __restrict__ Ci,
    int M, int N, int K, int lda, int ldb, int ldc,
    int aZ, long bZ, int cZ) {
  Ar += (long)blockIdx.z * aZ;
  Ai += (long)blockIdx.z * aZ;
  Bc += (long)blockIdx.z * bZ;
  Cr += (long)blockIdx.z * cZ;
  Ci += (long)blockIdx.z * cZ;

  __shared__ float Asr[GM][AP], Asi[GM][AP];
  __shared__ float Bsr[GK][BP], Bsi[GK][BP];

  const int tid = threadIdx.x;
  const int wave = tid >> 5, lane = tid & 31;
  const int lo = lane & 15, hi = lane >> 4;
  const int m0 = blockIdx.y * GM, n0 = blockIdx.x * GN;
  const int r0 = wave * 16;

  v8f vzero = {0.f, 0.f, 0.f, 0.f, 0.f, 0.f, 0.f, 0.f};
  v8f cr[4], ci[4];
#pragma unroll
  for (int t = 0; t < 4; ++t) { cr[t] = vzero; ci[t] = vzero; }

  for (int k0 = 0; k0 < K; k0 += GK) {
    // stage A tile (128x16 per plane, 8 elements / thread)
#pragma unroll
    for (int it = 0; it < 8; ++it) {
      int idx = tid + it * 256;
      int r = idx >> 4, c = idx & 15;
      int gm = m0 + r, gk = k0 + c;
      bool ok = (gm < M) && (gk < K);
      Asr[r][c] = ok ? Ar[(long)gm * lda + gk] : 0.f;
      Asi[r][c] = ok ? Ai[(long)gm * lda + gk] : 0.f;
    }
    // stage B tile (16x64 complex, de-interleave, 4 elements / thread)
#pragma unroll
    for (int it = 0; it < 4; ++it) {
      int idx = tid + it * 256;
      int r = idx >> 6, c = idx & 63;
      int gk = k0 + r, gn = n0 + c;
      float2 v; v.x = 0.f; v.y = 0.f;
      if (gk < K && gn < N) v = Bc[(long)gk * ldb + gn];
      Bsr[r][c] = v.x;
      Bsi[r][c] = v.y;
    }
    // prefetch next K-slab (global_prefetch_b8)
    if (k0 + GK < K) {
      int pr = m0 + (tid & 127);
      if (tid < 128 && pr < M) __builtin_prefetch(&Ar[(long)pr * lda + k0 + GK], 0, 0);
      if (tid >= 128 && (tid & 15) + k0 + GK < K)
        __builtin_prefetch(&Bc[(long)(k0 + GK + (tid & 15)) * ldb + n0], 0, 0);
    }
    __syncthreads();
#pragma unroll
    for (int kk = 0; kk < GK; kk += 4) {
      v2f ar, ai_, ain;
      ar.x  = Asr[r0 + lo][kk + 2 * hi];
      ar.y  = Asr[r0 + lo][kk + 2 * hi + 1];
      ai_.x = Asi[r0 + lo][kk + 2 * hi];
      ai_.y = Asi[r0 + lo][kk + 2 * hi + 1];
      ain = -ai_;
#pragma unroll
      for (int t = 0; t < 4; ++t) {
        v2f br, bi;
        br.x = Bsr[kk + 2 * hi][t * 16 + lo];
        br.y = Bsr[kk + 2 * hi + 1][t * 16 + lo];
        bi.x = Bsi[kk + 2 * hi][t * 16 + lo];
        bi.y = Bsi[kk + 2 * hi + 1][t * 16 + lo];
        cr[t] = wmma4(ar, br, cr[t]);   // + Ar*Br
        cr[t] = wmma4(ain, bi, cr[t]);  // - Ai*Bi
        ci[t] = wmma4(ar, bi, ci[t]);   // + Ar*Bi
        ci[t] = wmma4(ai_, br, ci[t]);  // + Ai*Br
      }
    }
    __syncthreads();
  }
#pragma unroll
  for (int t = 0; t < 4; ++t)
#pragma unroll
    for (int j = 0; j < 8; ++j) {
      int gm = m0 + r0 + j + hi * 8;
      int gn = n0 + t * 16 + lo;
      if (gm < M && gn < N) {
        Cr[(long)gm * ldc + gn] = cr[t][j];
        Ci[(long)gm * ldc + gn] = ci[t][j];
      }
    }
}

// ----------------------------- real GEMM (WMMA) ----------------------------
__global__ __launch_bounds__(256) void rgemm_kernel(
    const float* __restrict__ A, const float* __restrict__ Bm,
    float* __restrict__ C, int M, int N, int K, int lda, int ldb, int ldc) {
  __shared__ float As[GM][AP];
  __shared__ float Bs[GK][BP];

  const int tid = threadIdx.x;
  const int wave = tid >> 5, lane = tid & 31;
  const int lo = lane & 15, hi = lane >> 4;
  const int m0 = blockIdx.y * GM, n0 = blockIdx.x * GN;
  const int r0 = wave * 16;

  v8f vzero = {0.f, 0.f, 0.f, 0.f, 0.f, 0.f, 0.f, 0.f};
  v8f cc[4];
#pragma unroll
  for (int t = 0; t < 4; ++t) cc[t] = vzero;

  for (int k0 = 0; k0 < K; k0 += GK) {
#pragma unroll
    for (int it = 0; it < 8; ++it) {
      int idx = tid + it * 256;
      int r = idx >> 4, c = idx & 15;
      int gm = m0 + r, gk = k0 + c;
      As[r][c] = (gm < M && gk < K) ? A[(long)gm * lda + gk] : 0.f;
    }
#pragma unroll
    for (int it = 0; it < 4; ++it) {
      int idx = tid + it * 256;
      int r = idx >> 6, c = idx & 63;
      int gk = k0 + r, gn = n0 + c;
      Bs[r][c] = (gk < K && gn < N) ? Bm[(long)gk * ldb + gn] : 0.f;
    }
    if (k0 + GK < K && tid < 128 && m0 + tid < M)
      __builtin_prefetch(&A[(long)(m0 + tid) * lda + k0 + GK], 0, 0);
    __syncthreads();
#pragma unroll
    for (int kk = 0; kk < GK; kk += 4) {
      v2f a;
      a.x = As[r0 + lo][kk + 2 * hi];
      a.y = As[r0 + lo][kk + 2 * hi + 1];
#pragma unroll
      for (int t = 0; t < 4; ++t) {
        v2f b;
        b.x = Bs[kk + 2 * hi][t * 16 + lo];
        b.y = Bs[kk + 2 * hi + 1][t * 16 + lo];
        cc[t] = wmma4(a, b, cc[t]);
      }
    }
    __syncthreads();
  }
#pragma unroll
  for (int t = 0; t < 4; ++t)
#pragma unroll
    for (int j = 0; j < 8; ++j) {
      int gm = m0 + r0 + j + hi * 8;
      int gn = n0 + t * 16 + lo;
      if (gm < M && gn < N) C[(long)gm * ldc + gn] = cc[t][j];
    }
}

// ------------------------- fused retention (WMMA) --------------------------
// Per block: (b, h, 64 query rows). A = Q K^T (contract d=128) -> decay mask
// -> LDS relayout -> Y += A V (contract m=16).
#define RW  4     // waves per block
#define KP  132   // K/V LDS row pitch
#define QP  132   // Q LDS row pitch
#define AWP 20    // A-tile scratch pitch

__global__ __launch_bounds__(128) void retention_kernel(
    const float* __restrict__ Qgr, const float* __restrict__ Qgi,
    const float* __restrict__ Kgr, const float* __restrict__ Kgi,
    const float* __restrict__ Vgr, const float* __restrict__ Vgi,
    float* __restrict__ Yor, float* __restrict__ Yoi) {
  extern __shared__ float smem[];
  float* Qsr = smem;                    // RW*16 x QP
  float* Qsi = Qsr + RW * 16 * QP;
  float* Ksr = Qsi + RW * 16 * QP;      // 16 x KP
  float* Ksi = Ksr + 16 * KP;
  float* Vsr = Ksi + 16 * KP;
  float* Vsi = Vsr + 16 * KP;
  float* Awr = Vsi + 16 * KP;           // RW*16 x AWP
  float* Awi = Awr + RW * 16 * AWP;

  const int tid = threadIdx.x;
  const int wave = tid >> 5, lane = tid & 31;
  const int lo = lane & 15, hi = lane >> 4;
  const int b = blockIdx.y / HEADS_;
  const int h = blockIdx.y % HEADS_;
  const int n0 = blockIdx.x * (RW * 16);

  const float t0 = logf(1.f / 32.f), t1 = logf(1.f / 512.f);
  const float gamma = 1.f - __expf(t0 + (float)h * (t1 - t0) / 7.f);
  const float logg = logf(gamma);

  const long rowBase = (long)b * S_;
  const int colBase = h * HD_;

  // stage Q (64 rows x 128, both planes)
  for (int idx = tid; idx < RW * 16 * HD_; idx += 128) {
    int r = idx >> 7;
    int d = idx & 127;
    long g = (rowBase + n0 + r) * H_ + colBase + d;
    Qsr[r * QP + d] = Qgr[g];
    Qsi[r * QP + d] = Qgi[g];
  }
  __syncthreads();

  v8f vzero = {0.f, 0.f, 0.f, 0.f, 0.f, 0.f, 0.f, 0.f};
  v8f yr[8], yi[8];
#pragma unroll
  for (int t = 0; t < 8; ++t) { yr[t] = vzero; yi[t] = vzero; }

  const int mTiles = blockIdx.x * 4 + 4;   // causal: m-tile start <= n0+63
  for (int mt = 0; mt < mTiles; ++mt) {
    const int m0 = mt * 16;
    __syncthreads();  // previous iteration's V readers done
    for (int idx = tid; idx < 16 * HD_; idx += 128) {
      int r = idx >> 7, d = idx & 127;
      long g = (rowBase + m0 + r) * H_ + colBase + d;
      Ksr[r * KP + d] = Kgr[g];
      Ksi[r * KP + d] = Kgi[g];
      Vsr[r * KP + d] = Vgr[g];
      Vsi[r * KP + d] = Vgi[g];
    }
    __syncthreads();

    // ---- A(16x16) = Q(16x128) * K^T(128x16), complex ----
    v8f arAcc = vzero, aiAcc = vzero;
#pragma unroll
    for (int dk = 0; dk < HD_; dk += 4) {
      v2f qr, qi, qin, kr, ki;
      int qrow = (wave * 16 + lo) * QP + dk + 2 * hi;
      qr.x = Qsr[qrow]; qr.y = Qsr[qrow + 1];
      qi.x = Qsi[qrow]; qi.y = Qsi[qrow + 1];
      qin = -qi;
      int krow = lo * KP + dk + 2 * hi;   // B[k=d][n=m] = K[m][d]
      kr.x = Ksr[krow]; kr.y = Ksr[krow + 1];
      ki.x = Ksi[krow]; ki.y = Ksi[krow + 1];
      arAcc = wmma4(qr, kr, arAcc);
      arAcc = wmma4(qin, ki, arAcc);
      aiAcc = wmma4(qr, ki, aiAcc);
      aiAcc = wmma4(qi, kr, aiAcc);
    }
    // ---- decay mask + write A tile to per-wave LDS scratch ----
#pragma unroll
    for (int j = 0; j < 8; ++j) {
      int nl = j + hi * 8;
      int n = n0 + wave * 16 + nl;
      int m = m0 + lo;
      int diff = n - m;
      float w = (diff >= 0) ? __expf((float)diff * logg) : 0.f;
      Awr[(wave * 16 + nl) * AWP + lo] = arAcc[j] * w;
      Awi[(wave * 16 + nl) * AWP + lo] = aiAcc[j] * w;
    }
    __syncthreads();
    // ---- Y(16x128) += A(16x16) * V(16x128), complex ----
#pragma unroll
    for (int mk = 0; mk < 16; mk += 4) {
      v2f aR, aI, aIn;
      int arow = (wave * 16 + lo) * AWP + mk + 2 * hi;
      aR.x = Awr[arow]; aR.y = Awr[arow + 1];
      aI.x = Awi[arow]; aI.y = Awi[arow + 1];
      aIn = -aI;
#pragma unroll
      for (int dt = 0; dt < 8; ++dt) {
        v2f vr, vi;
        int vrow = (mk + 2 * hi) * KP + dt * 16 + lo;
        vr.x = Vsr[vrow]; vr.y = Vsr[vrow + KP];
        vi.x = Vsi[vrow]; vi.y = Vsi[vrow + KP];
        yr[dt] = wmma4(aR, vr, yr[dt]);
        yr[dt] = wmma4(aIn, vi, yr[dt]);
        yi[dt] = wmma4(aR, vi, yi[dt]);
        yi[dt] = wmma4(aI, vr, yi[dt]);
      }
    }
  }
  // store Y
#pragma unroll
  for (int dt = 0; dt < 8; ++dt)
#pragma unroll
    for (int j = 0; j < 8; ++j) {
      int n = n0 + wave * 16 + j + hi * 8;
      long g = (rowBase + n) * H_ + colBase + dt * 16 + lo;
      Yor[g] = yr[dt][j];
      Yoi[g] = yi[dt][j];
    }
}

// ---------------------------------------------------------------------------
extern "C" void kernel_launch(void* const* d_in, const int* in_sizes, int n_in,
                              void* d_out, int out_size, void* d_ws, size_t ws_size,
                              hipStream_t stream) {
  const int*    ids   = (const int*)d_in[0];
  const float*  emb   = (const float*)d_in[1];
  const float2* ln_g  = (const float2*)d_in[2];
  const float2* ln_b  = (const float2*)d_in[3];
  const float2* WQ    = (const float2*)d_in[4];
  const float2* WK    = (const float2*)d_in[5];
  const float2* WV    = (const float2*)d_in[6];
  const float2* WG    = (const float2*)d_in[7];
  const float2* WO    = (const float2*)d_in[8];
  const float2* gn_g  = (const float2*)d_in[9];
  const float2* gn_b  = (const float2*)d_in[10];
  const float2* W1    = (const float2*)d_in[11];
  const float2* W2    = (const float2*)d_in[12];
  const float*  proj  = (const float*)d_in[13];
  float* out = (float*)d_out;

  const long PLANE = (long)MROWS_ * H_;   // 2,097,152 floats
  if (ws_size < (size_t)16 * PLANE * sizeof(float)) return;
  float* a = (float*)d_ws;
  float *Xr  = a,             *Xi  = a + PLANE;
  float *Xnr = a + 2 * PLANE, *Xni = a + 3 * PLANE;
  float *Qr  = a + 4 * PLANE, *Qi  = a + 5 * PLANE;
  float *Kr  = a + 6 * PLANE, *Ki  = a + 7 * PLANE;
  float *Vr  = a + 8 * PLANE, *Vi  = a + 9 * PLANE;
  float *Yhr = a + 10 * PLANE, *Yhi = a + 11 * PLANE;
  float *Gr  = a + 12 * PLANE, *Gi  = a + 13 * PLANE;
  float *Yr2 = a + 14 * PLANE, *Yi2 = a + 15 * PLANE;
  // FFN hidden overlays dead Q/K planes (4 planes = 2048x2048 x2)
  float *h1r = a + 4 * PLANE;   // spans Qr,Qi
  float *h1i = a + 6 * PLANE;   // spans Kr,Ki

  const int nBH = MROWS_ * H_;
  const int nBF = MROWS_ * F_;
  const size_t retSmem =
      (size_t)(2 * RW * 16 * QP + 4 * 16 * KP + 2 * RW * 16 * AWP) * sizeof(float);

  embed_kernel<<<(nBH + 255) / 256, 256, 0, stream>>>(ids, emb, Xr, Xi, nBH);

  for (int l = 0; l < L_; ++l) {
    // pre-norm
    cnorm_kernel<<<MROWS_, 256, 0, stream>>>(Xr, Xi, ln_g, ln_b, Xnr, Xni, H_);
    // QKV per-head projections (batched over heads via blockIdx.z)
    dim3 gQ(HD_ / GN, MROWS_ / GM, HEADS_);
    const long wqkvL = (long)HEADS_ * HD_ * HD_;
    cgemm_kernel<<<gQ, 256, 0, stream>>>(Xnr, Xni, WQ + (long)l * wqkvL, Qr, Qi,
        MROWS_, HD_, HD_, H_, HD_, H_, HD_, (long)HD_ * HD_, HD_);
    cgemm_kernel<<<gQ, 256, 0, stream>>>(Xnr, Xni, WK + (long)l * wqkvL, Kr, Ki,
        MROWS_, HD_, HD_, H_, HD_, H_, HD_, (long)HD_ * HD_, HD_);
    cgemm_kernel<<<gQ, 256, 0, stream>>>(Xnr, Xni, WV + (long)l * wqkvL, Vr, Vi,
        MROWS_, HD_, HD_, H_, HD_, H_, HD_, (long)HD_ * HD_, HD_);
    // fused causal retention
    dim3 gR(S_ / (RW * 16), B_ * HEADS_);
    retention_kernel<<<gR, 128, retSmem, stream>>>(Qr, Qi, Kr, Ki, Vr, Vi, Yhr, Yhi);
    // per-head group norm + affine -> Yg (reuse V planes)
    gnorm_kernel<<<MROWS_ * HEADS_, 128, 0, stream>>>(Yhr, Yhi, gn_g, gn_b, Vr, Vi);
    // gate projection G = Xn @ WG
    dim3 gG(H_ / GN, MROWS_ / GM);
    cgemm_kernel<<<gG, 256, 0, stream>>>(Xnr, Xni, WG + (long)l * H_ * H_, Gr, Gi,
        MROWS_, H_, H_, H_, H_, H_, 0, 0, 0);
    // R = G * sig(G) * Yg (in-place into G)
    gate_kernel<<<(nBH + 255) / 256, 256, 0, stream>>>(Gr, Gi, Vr, Vi, Gr, Gi, nBH);
    // ret = R @ WO (into Yh planes)
    cgemm_kernel<<<gG, 256, 0, stream>>>(Gr, Gi, WO + (long)l * H_ * H_, Yhr, Yhi,
        MROWS_, H_, H_, H_, H_, H_, 0, 0, 0);
    // Y = ret + X
    cadd_kernel<<<(nBH + 255) / 256, 256, 0, stream>>>(Yhr, Yhi, Xr, Xi, Yr2, Yi2, nBH);
    // FFN pre-norm
    cnorm_kernel<<<MROWS_, 256, 0, stream>>>(Yr2, Yi2, ln_g, ln_b, Xnr, Xni, H_);
    // h1 = Yn @ W1
    dim3 g1(F_ / GN, MROWS_ / GM);
    cgemm_kernel<<<g1, 256, 0, stream>>>(Xnr, Xni, W1 + (long)l * H_ * F_, h1r, h1i,
        MROWS_, F_, H_, H_, F_, F_, 0, 0, 0);
    // gelu (sigmoid approx) in place
    cgelu_kernel<<<(nBF + 255) / 256, 256, 0, stream>>>(h1r, h1i, nBF);
    // ffn out = h1g @ W2 (into G planes)
    cgemm_kernel<<<gG, 256, 0, stream>>>(h1r, h1i, W2 + (long)l * F_ * H_, Gr, Gi,
        MROWS_, H_, F_, F_, H_, H_, 0, 0, 0);
    // X = ffn + Y
    cadd_kernel<<<(nBH + 255) / 256, 256, 0, stream>>>(Gr, Gi, Yr2, Yi2, Xr, Xi, nBH);
  }
  // logits = real(X @ proj) = Xr @ proj
  dim3 gP(V_ / GN, MROWS_ / GM);
  rgemm_kernel<<<gP, 256, 0, stream>>>(Xr, proj, out, MROWS_, V_, H_, H_, V_, V_);
}